// DistributedMemory_82514911690790
// MI455X (gfx1250) — compile-verified
//
#include <hip/hip_runtime.h>

typedef __attribute__((ext_vector_type(2))) float v2f;
typedef __attribute__((ext_vector_type(4))) float v4f;
typedef __attribute__((ext_vector_type(8))) float v8f;

#define VEC_DIM 128
#define CTX 8
#define NSAMP 10
#define NWORDS 100000
#define WAVES_PER_BLOCK 8
#define THREADS (WAVES_PER_BLOCK * 32)

// One wave32 per batch element b.
//  Phase 1: inputs[b,:] = P[doc[b],:] + sum_c W[ctx[b,c],:]   (float4 coalesced gathers)
//           staged into a 512B wave-private LDS slice.
//  Phase 2: result[b, 0..9] via V_WMMA_F32_16X16X4_F32, 32 chunks of K=4 over D=128.
//    A[m,k] = inputs[b, k0+k]  (broadcast over rows m; per ISA 16x4 A layout this is
//             lane-local {in[d0], in[d0+1]}, d0 = k0 + (lane<16 ? 0 : 2))
//    B[k,n] = outputs[k0+k, sample[b,n]]  (n = lane&15, samples padded 10->16)
//    => every row of D equals result[b, 0..15]; lane n of acc[0] = result[b,n].
__global__ __launch_bounds__(THREADS) void pvdm_score_kernel(
    const int* __restrict__ doc_ids,
    const int* __restrict__ context_ids,
    const int* __restrict__ sample_ids,
    const float* __restrict__ P,
    const float* __restrict__ W,
    const float* __restrict__ O,   // outputs: [VEC_DIM, NWORDS] row-major
    float* __restrict__ out,       // [B, NSAMP]
    int B)
{
    __shared__ __align__(16) float s_in[WAVES_PER_BLOCK][VEC_DIM];

    const int lane = threadIdx.x & 31;
    const int wv   = threadIdx.x >> 5;
    int b = blockIdx.x * WAVES_PER_BLOCK + wv;
    b = (b < B) ? b : (B - 1);          // clamp, not branch: EXEC stays all-1s

    // ---- Phase 1: build inputs[b, 4*lane .. 4*lane+3] ----
    const int doc = doc_ids[b];
    v4f acc = *(const v4f*)(P + (size_t)doc * VEC_DIM + 4 * lane);
#pragma unroll
    for (int cc = 0; cc < CTX; ++cc) {
        const int wid = context_ids[b * CTX + cc];
        acc += *(const v4f*)(W + (size_t)wid * VEC_DIM + 4 * lane);
    }
    *(v4f*)(&s_in[wv][4 * lane]) = acc;
    __syncthreads();

    // ---- Phase 2: WMMA-accumulated dot products ----
    const int n = lane & 15;                       // sample slot (valid for n < NSAMP)
    int col = 0;
    if (n < NSAMP) col = sample_ids[b * NSAMP + n];
    const float* colp = O + col;                   // column base of outputs
    const int kk = (lane >> 4) << 1;               // 0 for lanes 0-15, 2 for lanes 16-31

    v8f c = {0.f, 0.f, 0.f, 0.f, 0.f, 0.f, 0.f, 0.f};
#pragma unroll 4
    for (int k0 = 0; k0 < VEC_DIM; k0 += 4) {
        const int d0 = k0 + kk;                    // even -> 8B-aligned LDS read
        v2f a = *(const v2f*)(&s_in[wv][d0]);      // A: {in[d0], in[d0+1]}
        v2f bb;
        bb.x = colp[(size_t)d0 * NWORDS];          // B: outputs[d0,  col]
        bb.y = colp[(size_t)(d0 + 1) * NWORDS];    //    outputs[d0+1,col]
        c = __builtin_amdgcn_wmma_f32_16x16x4_f32(
                /*neg_a=*/false, a, /*neg_b=*/false, bb,
                /*c_mod=*/(short)0, c, /*reuse_a=*/false, /*reuse_b=*/false);
    }

    // acc VGPR0: lanes 0-15 hold D[M=0, N=lane] = result[b, lane]
    if (lane < NSAMP)
        out[b * NSAMP + lane] = c[0];
}

extern "C" void kernel_launch(void* const* d_in, const int* in_sizes, int n_in,
                              void* d_out, int out_size, void* d_ws, size_t ws_size,
                              hipStream_t stream) {
    const int*   doc_ids     = (const int*)d_in[0];
    const int*   context_ids = (const int*)d_in[1];
    const int*   sample_ids  = (const int*)d_in[2];
    const float* P           = (const float*)d_in[3];
    const float* W           = (const float*)d_in[4];
    const float* O           = (const float*)d_in[5];
    float*       out         = (float*)d_out;

    const int B = in_sizes[0];
    const int blocks = (B + WAVES_PER_BLOCK - 1) / WAVES_PER_BLOCK;
    pvdm_score_kernel<<<blocks, THREADS, 0, stream>>>(
        doc_ids, context_ids, sample_ids, P, W, O, out, B);
}